// DynamicAdjacency_5540507811924
// MI455X (gfx1250) — compile-verified
//
#include <hip/hip_runtime.h>
#include <cmath>

typedef float v2f __attribute__((ext_vector_type(2)));
typedef float v8f __attribute__((ext_vector_type(8)));

#define B_ 4
#define N_ 4096
#define D_ 512
#define K_ 32

#define TILE 128
#define KC 32
#define LDSS 36  // padded LDS stride (floats): 144B rows keep 16B alignment, avoid bank conflicts

// ---------------- 1) L2 normalize rows ----------------
__global__ __launch_bounds__(128) void normalize_kernel(const float* __restrict__ x,
                                                        float* __restrict__ xn) {
  const int row = blockIdx.x;          // 0..B*N-1
  const int tid = threadIdx.x;         // 0..127
  const float4* __restrict__ xin = (const float4*)(x + (size_t)row * D_);
  float4 v = xin[tid];                 // 128 threads * 4 floats = 512 = D
  float ss = v.x * v.x + v.y * v.y + v.z * v.z + v.w * v.w;
  // wave32 butterfly reduce
  #pragma unroll
  for (int off = 16; off > 0; off >>= 1) ss += __shfl_down(ss, off, 32);
  __shared__ float wsum[4];
  if ((tid & 31) == 0) wsum[tid >> 5] = ss;
  __syncthreads();
  float total = wsum[0] + wsum[1] + wsum[2] + wsum[3];
  float inv = 1.0f / fmaxf(sqrtf(total), 1e-12f);
  float4 o;
  o.x = v.x * inv; o.y = v.y * inv; o.z = v.z * inv; o.w = v.w * inv;
  ((float4*)(xn + (size_t)row * D_))[tid] = o;
}

// ---------------- 2) sim = xn @ xn^T via V_WMMA_F32_16X16X4_F32 ----------------
__global__ __launch_bounds__(256) void gemm_sim_kernel(const float* __restrict__ xn,
                                                       float* __restrict__ sim) {
  const int b = blockIdx.y;
  const int tilesPerSide = N_ / TILE;        // 32
  const int tr = blockIdx.x / tilesPerSide;  // row tile
  const int tc = blockIdx.x % tilesPerSide;  // col tile
  const int row0 = tr * TILE, col0 = tc * TILE;
  const float* __restrict__ Xb = xn + (size_t)b * N_ * D_;

  __shared__ __align__(16) float As[TILE][LDSS];
  __shared__ __align__(16) float Bs[TILE][LDSS];

  const int tid  = threadIdx.x;
  const int lane = tid & 31;
  const int wave = tid >> 5;       // 0..7: wave owns rows [wave*16, wave*16+16)
  const int half = lane >> 4;      // 0/1  (K pair select per ISA A/B layout)
  const int l16  = lane & 15;      // M (A) / N (B) index

  v8f acc[8] = {};                 // 8 col sub-tiles of 16x16 f32 accum

  for (int kc = 0; kc < D_; kc += KC) {
    // Stage A (rows) and B (cols) K-chunks: 256 threads x (1 float4 each side) x 4 passes
    const int r0 = tid >> 3;           // 0..31
    const int kq = (tid & 7) << 2;     // 0,4,...,28
    #pragma unroll
    for (int p = 0; p < 4; ++p) {
      const int r = r0 + p * 32;
      float4 av = *(const float4*)(Xb + (size_t)(row0 + r) * D_ + kc + kq);
      *(float4*)&As[r][kq] = av;
      float4 bv = *(const float4*)(Xb + (size_t)(col0 + r) * D_ + kc + kq);
      *(float4*)&Bs[r][kq] = bv;
    }
    __syncthreads();

    #pragma unroll
    for (int k0 = 0; k0 < KC; k0 += 4) {
      // A 16x4 fragment: lanes 0-15 -> {K0,K1}, lanes 16-31 -> {K2,K3}
      v2f a;
      a.x = As[wave * 16 + l16][k0 + 2 * half];
      a.y = As[wave * 16 + l16][k0 + 2 * half + 1];
      // Batch all 8 B fragments first so the LDS clause issues once and the
      // 8 WMMAs then run back-to-back (one s_wait_dscnt instead of eight).
      v2f bf[8];
      #pragma unroll
      for (int c = 0; c < 8; ++c) {
        bf[c].x = Bs[c * 16 + l16][k0 + 2 * half];
        bf[c].y = Bs[c * 16 + l16][k0 + 2 * half + 1];
      }
      #pragma unroll
      for (int c = 0; c < 8; ++c) {
        acc[c] = __builtin_amdgcn_wmma_f32_16x16x4_f32(
            /*neg_a=*/false, a, /*neg_b=*/false, bf[c],
            /*c_mod=*/(short)0, acc[c], /*reuse_a=*/false, /*reuse_b=*/false);
      }
    }
    __syncthreads();
  }

  // C/D layout: VGPR r, lanes 0-15 -> (M=r, N=lane); lanes 16-31 -> (M=8+r, N=lane-16)
  float* __restrict__ simb = sim + (size_t)b * N_ * N_;
  #pragma unroll
  for (int c = 0; c < 8; ++c) {
    const int col = col0 + c * 16 + l16;
    #pragma unroll
    for (int r = 0; r < 8; ++r) {
      const int row = row0 + wave * 16 + half * 8 + r;
      simb[(size_t)row * N_ + col] = acc[c][r];
    }
  }
}

// ---------------- 3) exact row-wise top-32 (lowest-index tie-break) ----------------
__global__ __launch_bounds__(256) void topk_kernel(const float* __restrict__ sim,
                                                   float* __restrict__ tv,
                                                   int* __restrict__ ti) {
  const int row = blockIdx.x;      // 0..B*N-1
  const int tid = threadIdx.x;
  __shared__ __align__(16) float srow[N_];   // 16KB row cache
  __shared__ float rv[256];
  __shared__ int   ri[256];

  const float4* __restrict__ p4 = (const float4*)(sim + (size_t)row * N_);
  float4* s4 = (float4*)srow;
  for (int i = tid; i < N_ / 4; i += 256) s4[i] = p4[i];
  __syncthreads();

  for (int it = 0; it < K_; ++it) {
    float bestv = -INFINITY;
    int   besti = 0x7fffffff;
    for (int j = tid; j < N_; j += 256) {   // ascending j + strict '>' keeps lowest index
      float v = srow[j];
      if (v > bestv) { bestv = v; besti = j; }
    }
    rv[tid] = bestv; ri[tid] = besti;
    __syncthreads();
    #pragma unroll
    for (int s = 128; s > 0; s >>= 1) {
      if (tid < s) {
        float ov = rv[tid + s]; int oi = ri[tid + s];
        if (ov > rv[tid] || (ov == rv[tid] && oi < ri[tid])) { rv[tid] = ov; ri[tid] = oi; }
      }
      __syncthreads();
    }
    if (tid == 0) {
      tv[(size_t)row * K_ + it] = rv[0];
      ti[(size_t)row * K_ + it] = ri[0];
      srow[ri[0]] = -INFINITY;     // knock out winner
    }
    __syncthreads();
  }
}

// ---------------- 4) zero output ----------------
__global__ void zero_kernel(float4* __restrict__ out, long long n4) {
  long long i = (long long)blockIdx.x * blockDim.x + threadIdx.x;
  const long long stride = (long long)gridDim.x * blockDim.x;
  const float4 z = {0.f, 0.f, 0.f, 0.f};
  for (; i < n4; i += stride) out[i] = z;
}

// ---------------- 5) symmetric scatter: out = (mask + mask^T)/2 ----------------
__global__ void scatter_kernel(const float* __restrict__ tv, const int* __restrict__ ti,
                               float* __restrict__ out) {
  const int g = blockIdx.x * blockDim.x + threadIdx.x;  // 0..B*N*K-1
  if (g >= B_ * N_ * K_) return;
  const int row = (g / K_) % N_;
  const int b   = g / (K_ * N_);
  const float v = tv[g] * 0.5f;
  const int col = ti[g];
  float* ob = out + (size_t)b * N_ * N_;
  atomicAdd(&ob[(size_t)row * N_ + col], v);
  atomicAdd(&ob[(size_t)col * N_ + row], v);
}

extern "C" void kernel_launch(void* const* d_in, const int* in_sizes, int n_in,
                              void* d_out, int out_size, void* d_ws, size_t ws_size,
                              hipStream_t stream) {
  (void)in_sizes; (void)n_in; (void)out_size; (void)ws_size;
  const float* x = (const float*)d_in[0];
  float* out = (float*)d_out;

  // workspace layout: xn (32MB) | topk vals (2MB) | topk idx (2MB)
  float* xn = (float*)d_ws;
  float* tv = xn + (size_t)B_ * N_ * D_;
  int*   ti = (int*)(tv + (size_t)B_ * N_ * K_);
  float* sim = out;  // reuse d_out as dense sim scratch; fully overwritten later

  normalize_kernel<<<B_ * N_, 128, 0, stream>>>(x, xn);

  dim3 gGemm((N_ / TILE) * (N_ / TILE), B_);   // (1024, 4)
  gemm_sim_kernel<<<gGemm, 256, 0, stream>>>(xn, sim);

  topk_kernel<<<B_ * N_, 256, 0, stream>>>(sim, tv, ti);

  const long long n4 = (long long)B_ * N_ * N_ / 4;
  zero_kernel<<<4096, 256, 0, stream>>>((float4*)out, n4);

  const int nScatter = B_ * N_ * K_;
  scatter_kernel<<<(nScatter + 255) / 256, 256, 0, stream>>>(tv, ti, out);
}